// GaussianRasterizer_3977139716587
// MI455X (gfx1250) — compile-verified
//
#include <hip/hip_runtime.h>
#include <hip/hip_bf16.h>
#include <math.h>
#include <stdint.h>

typedef float v2f __attribute__((ext_vector_type(2)));
typedef float v8f __attribute__((ext_vector_type(8)));

#define TANF 0.57735026918962576f     // tan(FOV/2), FOV = 1.0471975512
#define SH_C0 0.28209479177387814f
#define LOWPASS 0.3f

// ---------------------------------------------------------------------------
// Kernel 1: per-Gaussian preprocessing
// ---------------------------------------------------------------------------
__global__ __launch_bounds__(256)
void gs_preprocess(const float* __restrict__ means,
                   const float* __restrict__ feats,
                   const float* __restrict__ opac,
                   const float* __restrict__ scales,
                   const float* __restrict__ rots,
                   const float* __restrict__ vm,   // 4x4 row-major
                   const float* __restrict__ pm,   // 4x4 row-major
                   const int*   __restrict__ pH,
                   const int*   __restrict__ pW,
                   float* __restrict__ upar,       // P*8: px,py,ca,cb,cc,op,0,0
                   float* __restrict__ ucol,       // P*4: r,g,b,0
                   float* __restrict__ keys,       // P
                   float* __restrict__ outRad,     // P
                   int P)
{
    int i = blockIdx.x * blockDim.x + threadIdx.x;
    if (i >= P) return;

    float W = (float)(*pW);
    float H = (float)(*pH);
    float fx = W / (2.0f * TANF);
    float fy = H / (2.0f * TANF);
    float lim = 1.3f * TANF;

    float mx = means[i*3+0], my = means[i*3+1], mz = means[i*3+2];

    // view transform
    float t0 = vm[0]*mx + vm[1]*my + vm[2]*mz + vm[3];
    float t1 = vm[4]*mx + vm[5]*my + vm[6]*mz + vm[7];
    float t2 = vm[8]*mx + vm[9]*my + vm[10]*mz + vm[11];
    float depth = t2;

    // projection
    float ph0 = pm[0]*mx + pm[1]*my + pm[2]*mz + pm[3];
    float ph1 = pm[4]*mx + pm[5]*my + pm[6]*mz + pm[7];
    float pww = pm[12]*mx + pm[13]*my + pm[14]*mz + pm[15];
    float invpw = 1.0f / (pww + 1e-7f);
    float px = ((ph0*invpw + 1.0f) * W - 1.0f) * 0.5f;
    float py = ((ph1*invpw + 1.0f) * H - 1.0f) * 0.5f;

    // quaternion -> rotation
    float qw = rots[i*4+0], qx = rots[i*4+1], qy = rots[i*4+2], qz = rots[i*4+3];
    float qinv = rsqrtf(qw*qw + qx*qx + qy*qy + qz*qz);
    qw *= qinv; qx *= qinv; qy *= qinv; qz *= qinv;
    float r00 = 1.f - 2.f*(qy*qy + qz*qz), r01 = 2.f*(qx*qy - qw*qz), r02 = 2.f*(qx*qz + qw*qy);
    float r10 = 2.f*(qx*qy + qw*qz), r11 = 1.f - 2.f*(qx*qx + qz*qz), r12 = 2.f*(qy*qz - qw*qx);
    float r20 = 2.f*(qx*qz - qw*qy), r21 = 2.f*(qy*qz + qw*qx), r22 = 1.f - 2.f*(qx*qx + qy*qy);

    float sx = scales[i*3+0], sy = scales[i*3+1], sz = scales[i*3+2];
    float m00=r00*sx, m01=r01*sy, m02=r02*sz;
    float m10=r10*sx, m11=r11*sy, m12=r12*sz;
    float m20=r20*sx, m21=r21*sy, m22=r22*sz;

    // sigma = M M^T (symmetric 3x3)
    float s00 = m00*m00 + m01*m01 + m02*m02;
    float s01 = m00*m10 + m01*m11 + m02*m12;
    float s02 = m00*m20 + m01*m21 + m02*m22;
    float s11 = m10*m10 + m11*m11 + m12*m12;
    float s12 = m10*m20 + m11*m21 + m12*m22;
    float s22 = m20*m20 + m21*m21 + m22*m22;

    float invz = 1.0f / t2;
    float tx = fminf(fmaxf(t0*invz, -lim), lim) * t2;
    float ty = fminf(fmaxf(t1*invz, -lim), lim) * t2;
    float j00 = fx * invz;
    float j02 = -fx * tx * invz * invz;
    float j11 = fy * invz;
    float j12 = -fy * ty * invz * invz;

    // Tm = J @ V[:3,:3]  (2x3)
    float Tm00 = j00*vm[0] + j02*vm[8];
    float Tm01 = j00*vm[1] + j02*vm[9];
    float Tm02 = j00*vm[2] + j02*vm[10];
    float Tm10 = j11*vm[4] + j12*vm[8];
    float Tm11 = j11*vm[5] + j12*vm[9];
    float Tm12 = j11*vm[6] + j12*vm[10];

    // U = Tm @ sigma (2x3)
    float u00 = Tm00*s00 + Tm01*s01 + Tm02*s02;
    float u01 = Tm00*s01 + Tm01*s11 + Tm02*s12;
    float u02 = Tm00*s02 + Tm01*s12 + Tm02*s22;
    float u10 = Tm10*s00 + Tm11*s01 + Tm12*s02;
    float u11 = Tm10*s01 + Tm11*s11 + Tm12*s12;
    float u12 = Tm10*s02 + Tm11*s12 + Tm12*s22;

    float cov00 = u00*Tm00 + u01*Tm01 + u02*Tm02;
    float cov01 = u00*Tm10 + u01*Tm11 + u02*Tm12;
    float cov11 = u10*Tm10 + u11*Tm11 + u12*Tm12;

    float a = cov00 + LOWPASS;
    float b = cov01;
    float c = cov11 + LOWPASS;
    float det = a*c - b*b;
    bool valid = (depth > 0.2f) && (det > 0.0f);
    float dets = valid ? det : 1.0f;
    float dinv = 1.0f / dets;
    float ca = c * dinv, cb = -b * dinv, cc = a * dinv;

    float mid = 0.5f * (a + c);
    float lam = mid + sqrtf(fmaxf(mid*mid - det, 0.1f));
    float rad = valid ? ceilf(3.0f * sqrtf(lam)) : 0.0f;

    float cr  = fmaxf(SH_C0 * feats[i*3+0] + 0.5f, 0.0f);
    float cg  = fmaxf(SH_C0 * feats[i*3+1] + 0.5f, 0.0f);
    float cbl = fmaxf(SH_C0 * feats[i*3+2] + 0.5f, 0.0f);
    float op  = valid ? opac[i] : 0.0f;

    upar[i*8+0] = px;  upar[i*8+1] = py;
    upar[i*8+2] = ca;  upar[i*8+3] = cb;
    upar[i*8+4] = cc;  upar[i*8+5] = op;
    upar[i*8+6] = 0.f; upar[i*8+7] = 0.f;
    ucol[i*4+0] = cr;  ucol[i*4+1] = cg;
    ucol[i*4+2] = cbl; ucol[i*4+3] = 0.f;
    keys[i]   = valid ? depth : __builtin_inff();
    outRad[i] = rad;
}

// ---------------------------------------------------------------------------
// Kernel 2: single-block bitonic sort by (depth, index) + scatter into sorted
// param / color arrays. Stable-argsort equivalent via lexicographic tiebreak.
// ---------------------------------------------------------------------------
#define SORT_N 2048

__global__ __launch_bounds__(1024)
void gs_sort_scatter(const float* __restrict__ keys,
                     const float* __restrict__ upar,
                     const float* __restrict__ ucol,
                     float* __restrict__ spar,
                     float* __restrict__ scol,
                     int P)
{
    __shared__ float sk[SORT_N];
    __shared__ int   si[SORT_N];
    int t = threadIdx.x;

    for (int e = t; e < SORT_N; e += 1024) {
        sk[e] = (e < P) ? keys[e] : __builtin_inff();
        si[e] = e;
    }
    __syncthreads();

    for (int k = 2; k <= SORT_N; k <<= 1) {
        for (int j = k >> 1; j > 0; j >>= 1) {
            for (int e = t; e < SORT_N; e += 1024) {
                int l = e ^ j;
                if (l > e) {
                    bool up = ((e & k) == 0);
                    float ke = sk[e], kl = sk[l];
                    int   ie = si[e], il = si[l];
                    bool gt = (ke > kl) || ((ke == kl) && (ie > il));
                    if (gt == up) {
                        sk[e] = kl; sk[l] = ke;
                        si[e] = il; si[l] = ie;
                    }
                }
            }
            __syncthreads();
        }
    }

    for (int e = t; e < P; e += 1024) {
        int g = si[e];
        const float4* up4 = (const float4*)upar;
        ((float4*)spar)[e*2+0] = up4[g*2+0];
        ((float4*)spar)[e*2+1] = up4[g*2+1];
        ((float4*)scol)[e]     = ((const float4*)ucol)[g];
    }
}

// ---------------------------------------------------------------------------
// Kernel 3: fused alpha compositing + WMMA color accumulation.
// Wave32: 16 pixels per wave (N dim), lane half h owns K slots {2h,2h+1}.
// C tile (16x16 f32): rows 0..2 = RGB channels, cols = pixels.
// Chunk staging uses GLOBAL_LOAD_ASYNC_TO_LDS_B128 (ASYNCcnt path).
// ---------------------------------------------------------------------------
__device__ __forceinline__
float eval_alpha(float4 pa, float4 pb, float gx, float gy)
{
    float dx = gx - pa.x;
    float dy = gy - pa.y;
    // power = -0.5*(ca*dx^2 + cc*dy^2) - cb*dx*dy
    float pw = fmaf(-0.5f * pa.z, dx*dx, fmaf(-0.5f * pb.x, dy*dy, -pa.w * dx * dy));
    float e  = __expf(fminf(pw, 0.0f));
    float al = fminf(0.99f, pb.y * e);
    return (pw > 0.0f || al < (1.0f / 255.0f)) ? 0.0f : al;
}

__device__ __forceinline__ void wait_asynccnt0()
{
#if __has_builtin(__builtin_amdgcn_s_wait_asynccnt)
    __builtin_amdgcn_s_wait_asynccnt(0);
#else
    asm volatile("s_wait_asynccnt 0x0" ::: "memory");
#endif
}

#define CHUNK 256

__global__ __launch_bounds__(256)
void gs_rasterize(const float4* __restrict__ spar,   // P*2 float4
                  const float4* __restrict__ scol,   // P   float4 (rgb0)
                  const int* __restrict__ pW,
                  float* __restrict__ outImg,        // 3*HW
                  int P, int HW)
{
    __shared__ float4 cp[CHUNK * 2];
    __shared__ float4 cc[CHUNK];
    const float* ccf = (const float*)cc;

    int W    = *pW;
    int tid  = threadIdx.x;
    int lane = tid & 31;
    int L    = lane & 15;     // pixel-within-tile (N)
    int h    = lane >> 4;     // K half: slots {2h, 2h+1}
    int wave = blockIdx.x * (blockDim.x >> 5) + (tid >> 5);
    int pix  = wave * 16 + L;
    int p    = (pix < HW) ? pix : 0;

    float gx = (float)(p % W);
    float gy = (float)(p / W);

    float T = 1.0f;
    v8f acc = {};

    int nChunk = (P + CHUNK - 1) / CHUNK;
    for (int cidx = 0; cidx < nChunk; ++cidx) {
        int base = cidx * CHUNK;
        int g    = base + tid;                    // blockDim.x == CHUNK
        bool chunkFull = (base + CHUNK) <= P;

        if (chunkFull) {
            // Async DMA: global -> LDS, no VGPR data, tracked by ASYNCcnt.
            // IOFFSET is added to BOTH global and LDS addresses, so the two
            // contiguous param beats share one address pair (offset 0 / 16).
            uint32_t ldsPar = (uint32_t)(uintptr_t)(&cp[tid * 2]);
            uint32_t ldsCol = (uint32_t)(uintptr_t)(&cc[tid]);
            uint64_t gaPar  = (uint64_t)(uintptr_t)(spar + (size_t)g * 2);
            uint64_t gaCol  = (uint64_t)(uintptr_t)(scol + (size_t)g);
            asm volatile(
                "global_load_async_to_lds_b128 %0, %1, off\n\t"
                "global_load_async_to_lds_b128 %0, %1, off offset:16\n\t"
                "global_load_async_to_lds_b128 %2, %3, off"
                :: "v"(ldsPar), "v"(gaPar), "v"(ldsCol), "v"(gaCol)
                : "memory");
            if (g + CHUNK < P)
                __builtin_prefetch(&spar[(size_t)(g + CHUNK) * 2], 0, 1);
            wait_asynccnt0();
        } else {
            // generic tail path (not taken when P % CHUNK == 0)
            float4 pa, pb, col;
            if (g < P) {
                pa = spar[g*2+0]; pb = spar[g*2+1]; col = scol[g];
            } else {
                pa = make_float4(0.f, 0.f, 0.f, 0.f);   // op=0 -> alpha=0
                pb = pa; col = pa;
            }
            cp[tid*2+0] = pa;
            cp[tid*2+1] = pb;
            cc[tid]     = col;
        }
        __syncthreads();

        #pragma unroll 4
        for (int g4 = 0; g4 < CHUNK; g4 += 4) {
            int k0 = g4 + 2*h;
            float4 qa = cp[k0*2+0], qb = cp[k0*2+1];
            float4 ra = cp[k0*2+2], rb = cp[k0*2+3];

            float a0  = eval_alpha(qa, qb, gx, gy);
            float a1  = eval_alpha(ra, rb, gx, gy);
            float om0 = 1.0f - a0, om1 = 1.0f - a1;
            float mloc = om0 * om1;                       // (1-a[2h])(1-a[2h+1])
            float moth = __shfl_xor(mloc, 16, 32);        // partner half product
            float pre  = h ? moth : 1.0f;                 // exclusive prefix for this half
            float tw   = T * pre;

            // B (4x16 weights): this lane's K slots, column = pixel L
            v2f Bv; Bv.x = tw * a0; Bv.y = tw * om0 * a1;

            // A (16x4 colors): row = L (channels 0..2, rest read padded 0)
            int ci = (L < 3) ? L : 3;                     // component 3 is 0.0
            v2f Av; Av.x = ccf[k0*4 + ci]; Av.y = ccf[(k0+1)*4 + ci];

            acc = __builtin_amdgcn_wmma_f32_16x16x4_f32(
                false, Av, false, Bv, (short)0, acc, false, false);

            T *= mloc * moth;                             // all 4 slots consumed
        }
        __syncthreads();
    }

    // C layout: lane VGPR r holds C[M = r + 8h, N = L]; channels live in h==0.
    if (h == 0 && pix < HW) {
        outImg[0*HW + pix] = acc[0] + T;
        outImg[1*HW + pix] = acc[1] + T;
        outImg[2*HW + pix] = acc[2] + T;
    }
}

// ---------------------------------------------------------------------------
extern "C" void kernel_launch(void* const* d_in, const int* in_sizes, int n_in,
                              void* d_out, int out_size, void* d_ws, size_t ws_size,
                              hipStream_t stream)
{
    const float* means  = (const float*)d_in[0];
    // d_in[1] = means_2d (unused)
    const float* feats  = (const float*)d_in[2];
    const float* opac   = (const float*)d_in[3];
    const float* scales = (const float*)d_in[4];
    const float* rots   = (const float*)d_in[5];
    const float* viewm  = (const float*)d_in[6];
    const float* projm  = (const float*)d_in[7];
    // d_in[8] = campos (unused)
    const int*   pH     = (const int*)d_in[9];
    const int*   pW     = (const int*)d_in[10];

    int P  = in_sizes[0] / 3;
    int HW = (out_size - P) / 3;

    float* outImg = (float*)d_out;
    float* outRad = (float*)d_out + 3 * HW;

    // workspace layout (floats)
    float* upar = (float*)d_ws;        // P*8
    float* ucol = upar + (size_t)P*8;  // P*4
    float* keys = ucol + (size_t)P*4;  // P
    float* spar = keys + (size_t)P;    // P*8
    float* scol = spar + (size_t)P*8;  // P*4

    gs_preprocess<<<(P + 255) / 256, 256, 0, stream>>>(
        means, feats, opac, scales, rots, viewm, projm, pH, pW,
        upar, ucol, keys, outRad, P);

    gs_sort_scatter<<<1, 1024, 0, stream>>>(keys, upar, ucol, spar, scol, P);

    int pxPerBlock = 128;   // 8 waves * 16 pixels
    gs_rasterize<<<(HW + pxPerBlock - 1) / pxPerBlock, 256, 0, stream>>>(
        (const float4*)spar, (const float4*)scol, pW, outImg, P, HW);
}